// GNNReranker_61091614818664
// MI455X (gfx1250) — compile-verified
//
#include <hip/hip_runtime.h>
#include <hip/hip_bf16.h>

typedef __attribute__((ext_vector_type(2))) float v2f;
typedef __attribute__((ext_vector_type(8))) float v8f;

#define HD 128      // hidden dim
#define TILE 16

// ---------------------------------------------------------------------------
// small utility kernels
// ---------------------------------------------------------------------------

__global__ void init_kernel(float* __restrict__ sumsq, float* __restrict__ deg, int n) {
    int i = blockIdx.x * blockDim.x + threadIdx.x;
    if (i == 0) sumsq[0] = 0.0f;
    if (i < n) deg[i] = 1.0f;              // self-loop weight contribution
}

// per-edge weight = sum of 8 features; accumulate global sum of squares
__global__ void ew_kernel(const float* __restrict__ ea, float* __restrict__ ew_raw,
                          float* __restrict__ sumsq, int E) {
    __shared__ float red[256];
    int e = blockIdx.x * blockDim.x + threadIdx.x;
    float s = 0.0f;
    if (e < E) {
        const float4* p = (const float4*)(ea + (size_t)e * 8);
        float4 a = p[0], b = p[1];
        s = ((a.x + a.y) + (a.z + a.w)) + ((b.x + b.y) + (b.z + b.w));
        ew_raw[e] = s;
    }
    red[threadIdx.x] = s * s;
    __syncthreads();
    for (int off = 128; off > 0; off >>= 1) {
        if (threadIdx.x < off) red[threadIdx.x] += red[threadIdx.x + off];
        __syncthreads();
    }
    if (threadIdx.x == 0) atomicAdd(sumsq, red[0]);
}

__global__ void scale_kernel(const float* __restrict__ sumsq, float* __restrict__ scale) {
    scale[0] = 1.0f / fmaxf(sqrtf(sumsq[0]), 1e-12f);
}

// scale edge weights in place and accumulate weighted in-degree at dst
__global__ void deg_kernel(const int* __restrict__ dst, float* __restrict__ ew,
                           const float* __restrict__ scale, float* __restrict__ deg, int E) {
    int e = blockIdx.x * blockDim.x + threadIdx.x;
    if (e < E) {
        float w = ew[e] * scale[0];
        ew[e] = w;                          // now fully normalized edge weight
        atomicAdd(&deg[dst[e]], w);
    }
}

// deg -> dinv in place
__global__ void dinv_kernel(float* __restrict__ deg, int n) {
    int i = blockIdx.x * blockDim.x + threadIdx.x;
    if (i < n) {
        float d = deg[i];
        deg[i] = (d > 0.0f) ? rsqrtf(d) : 0.0f;
    }
}

// ---------------------------------------------------------------------------
// fp32 WMMA GEMM:  out[M x 128] = act( A[M x K] @ W[128 x K]^T + bias )
// One workgroup (8 wave32) per 32-row stripe; wave w owns column tile w.
// Each wave computes TWO 16x16 tiles (rows row0..+15 and row0+16..+31) that
// share one B (weight) load per K-step: 3 loads per 2 WMMAs.
// V_WMMA_F32_16X16X4_F32 layouts:
//   A 16x4 : lane(l) -> row = l&15, K pair = (l>>4)*2 .. +1   (v2f)
//   B 4x16 : lane(l) -> col = l&15, K pair = (l>>4)*2 .. +1   (v2f) = W[col, k..k+1]
//   C/D    : vgpr r, lane l -> (row = r + 8*(l>>4), col = l&15)
// ---------------------------------------------------------------------------
__global__ __launch_bounds__(256)
void gemm_wmma_kernel(const float* __restrict__ A, const float* __restrict__ W,
                      const float* __restrict__ bias, float* __restrict__ out,
                      int M, int K, int do_relu) {
    const int lane = threadIdx.x & 31;
    const int wave = threadIdx.x >> 5;              // 0..7 -> output col tile
    const int row0 = blockIdx.x * (2 * TILE);
    const int col0 = wave * TILE;
    const int m    = lane & 15;                     // A row / B col within tile
    const int kk   = (lane >> 4) * 2;               // K sub-offset

    int ar0 = row0 + m;        if (ar0 >= M) ar0 = M - 1;   // clamp (stores guarded)
    int ar1 = row0 + TILE + m; if (ar1 >= M) ar1 = M - 1;
    const float* ap0 = A + (size_t)ar0 * K + kk;
    const float* ap1 = A + (size_t)ar1 * K + kk;
    const float* wp  = W + (size_t)(col0 + m) * K + kk;

    v8f acc0 = {};
    v8f acc1 = {};
#pragma unroll 4
    for (int k0 = 0; k0 < K; k0 += 4) {
        v2f b  = *(const v2f*)(wp  + k0);
        v2f a0 = *(const v2f*)(ap0 + k0);
        v2f a1 = *(const v2f*)(ap1 + k0);
        acc0 = __builtin_amdgcn_wmma_f32_16x16x4_f32(
            false, a0, false, b, (short)0, acc0, false, false);
        acc1 = __builtin_amdgcn_wmma_f32_16x16x4_f32(
            false, a1, false, b, (short)0, acc1, false, false);
    }

    const int n  = col0 + (lane & 15);
    const float bv = bias ? bias[n] : 0.0f;
    const int rbase = row0 + ((lane >> 4) << 3);
#pragma unroll
    for (int r = 0; r < 8; ++r) {
        int rowa = rbase + r;
        if (rowa < M) {
            float v = acc0[r] + bv;
            if (do_relu) v = fmaxf(v, 0.0f);
            out[(size_t)rowa * HD + n] = v;
        }
        int rowb = rbase + TILE + r;
        if (rowb < M) {
            float v = acc1[r] + bv;
            if (do_relu) v = fmaxf(v, 0.0f);
            out[(size_t)rowb * HD + n] = v;
        }
    }
}

// ---------------------------------------------------------------------------
// GCN scatter stage
// ---------------------------------------------------------------------------

// acc = xw * dinv^2  (self-loop term, initializes the accumulator)
__global__ void selfloop_kernel(const float* __restrict__ xw, const float* __restrict__ dinv,
                                float* __restrict__ acc, int n) {
    int idx = blockIdx.x * blockDim.x + threadIdx.x;
    if (idx < n * HD) {
        float d = dinv[idx >> 7];
        acc[idx] = xw[idx] * (d * d);
    }
}

// one wave32 per edge, each lane scatters 4 features with f32 atomics
__global__ void scatter_kernel(const int* __restrict__ src, const int* __restrict__ dst,
                               const float* __restrict__ ew, const float* __restrict__ dinv,
                               const float* __restrict__ xw, float* __restrict__ acc, int E) {
    int lane = threadIdx.x & 31;
    int e = (blockIdx.x * blockDim.x + threadIdx.x) >> 5;
    if (e >= E) return;
    int s = src[e], d = dst[e];
    float c = dinv[s] * ew[e] * dinv[d];
    float4 v = *(const float4*)(xw + (size_t)s * HD + lane * 4);
    float* p = acc + (size_t)d * HD + lane * 4;
    atomicAdd(p + 0, v.x * c);
    atomicAdd(p + 1, v.y * c);
    atomicAdd(p + 2, v.z * c);
    atomicAdd(p + 3, v.w * c);
}

// h = act(h + bias)  in place
__global__ void bias_act_kernel(float* __restrict__ h, const float* __restrict__ bias,
                                int n, int do_relu) {
    int idx = blockIdx.x * blockDim.x + threadIdx.x;
    if (idx < n * HD) {
        float v = h[idx] + bias[idx & (HD - 1)];
        if (do_relu) v = fmaxf(v, 0.0f);
        h[idx] = v;
    }
}

// ---------------------------------------------------------------------------

static inline size_t align256(size_t x) { return (x + 255) & ~(size_t)255; }

extern "C" void kernel_launch(void* const* d_in, const int* in_sizes, int n_in,
                              void* d_out, int out_size, void* d_ws, size_t ws_size,
                              hipStream_t stream) {
    const float* x     = (const float*)d_in[0];
    const int*   ei    = (const int*)  d_in[1];   // [2, E] int32
    const float* ea    = (const float*)d_in[2];   // [E, 8]
    const float* W_in  = (const float*)d_in[3];   // [128, 256]
    const float* b_in  = (const float*)d_in[4];
    const float* W1    = (const float*)d_in[5];   // [128, 128]
    const float* b1    = (const float*)d_in[6];
    const float* W2    = (const float*)d_in[7];
    const float* b2    = (const float*)d_in[8];
    const float* W_out = (const float*)d_in[9];
    const float* b_out = (const float*)d_in[10];

    const int E    = in_sizes[1] / 2;
    const int D_IN = in_sizes[3] / HD;            // 256
    const int N    = in_sizes[0] / D_IN;          // 100000
    const int* srcIdx = ei;
    const int* dstIdx = ei + E;

    // workspace layout
    char* ws = (char*)d_ws;
    size_t off = 0;
    float* sumsq  = (float*)(ws + off); off += 256;
    float* scale  = sumsq + 1;
    float* ew     = (float*)(ws + off); off += align256((size_t)E * 4);
    float* dinv   = (float*)(ws + off); off += align256((size_t)N * 4);   // deg -> dinv in place
    float* bufA   = (float*)(ws + off); off += (size_t)N * HD * 4;
    float* bufB   = (float*)(ws + off); off += (size_t)N * HD * 4;

    const int B256 = 256;
    dim3 blk(B256);
    int nBlkN  = (N + B256 - 1) / B256;
    int nBlkE  = (E + B256 - 1) / B256;
    int nBlkNH = (N * HD + B256 - 1) / B256;
    int nBlkEw = (E + 7) / 8;                     // one wave per edge
    int nBlkM  = (N + 2 * TILE - 1) / (2 * TILE); // 3125 row stripes (32 rows each)

    // edge weight / degree preprocessing
    init_kernel<<<nBlkN, blk, 0, stream>>>(sumsq, dinv, N);
    ew_kernel<<<nBlkE, blk, 0, stream>>>(ea, ew, sumsq, E);
    scale_kernel<<<1, 1, 0, stream>>>(sumsq, scale);
    deg_kernel<<<nBlkE, blk, 0, stream>>>(dstIdx, ew, scale, dinv, E);
    dinv_kernel<<<nBlkN, blk, 0, stream>>>(dinv, N);

    // h0 = relu(x @ W_in^T + b_in)                    -> bufA
    gemm_wmma_kernel<<<nBlkM, blk, 0, stream>>>(x, W_in, b_in, bufA, N, D_IN, 1);

    // GCN layer 1: xw = h0 @ W1^T                     -> bufB
    gemm_wmma_kernel<<<nBlkM, blk, 0, stream>>>(bufA, W1, nullptr, bufB, N, HD, 0);
    selfloop_kernel<<<nBlkNH, blk, 0, stream>>>(bufB, dinv, bufA, N);
    scatter_kernel<<<nBlkEw, blk, 0, stream>>>(srcIdx, dstIdx, ew, dinv, bufB, bufA, E);
    bias_act_kernel<<<nBlkNH, blk, 0, stream>>>(bufA, b1, N, 1);

    // GCN layer 2
    gemm_wmma_kernel<<<nBlkM, blk, 0, stream>>>(bufA, W2, nullptr, bufB, N, HD, 0);
    selfloop_kernel<<<nBlkNH, blk, 0, stream>>>(bufB, dinv, bufA, N);
    scatter_kernel<<<nBlkEw, blk, 0, stream>>>(srcIdx, dstIdx, ew, dinv, bufB, bufA, E);
    bias_act_kernel<<<nBlkNH, blk, 0, stream>>>(bufA, b2, N, 0);

    // out = h @ W_out^T + b_out                       -> d_out
    gemm_wmma_kernel<<<nBlkM, blk, 0, stream>>>(bufA, W_out, b_out, (float*)d_out, N, HD, 0);
}